// GAT_34179349742033
// MI455X (gfx1250) — compile-verified
//
#include <hip/hip_runtime.h>
#include <hip/hip_bf16.h>
#include <math.h>

#define GN   50000
#define GE   800000
#define FIN  128
#define HID  64
#define OUTC 40

typedef __attribute__((ext_vector_type(2))) float v2f;
typedef __attribute__((ext_vector_type(8))) float v8f;

// ---------------------------------------------------------------------------
// GEMM: C[N,M] = A[N,K] @ B[K, CT*16], fp32 WMMA 16x16x4.
// One wave per 16-row strip; the wave sweeps all CT column tiles so the A
// fragment is reused CT times per K-step. B must be padded to CT*16 columns
// (zero-filled) so the hot loop has NO divergence -> EXEC stays all-1s for
// every WMMA with no saveexec traffic. Stores are guarded (col < M) after
// the WMMA loop. grid = N/16 blocks, block = 32 (one wave32).
// ---------------------------------------------------------------------------
template <int CT>
__global__ void gat_gemm_wmma(const float* __restrict__ A,
                              const float* __restrict__ B,
                              float* __restrict__ C, int K, int M) {
  const int lane = threadIdx.x & 31;
  const int half = lane >> 4;              // 0: K pair {0,1}, 1: K pair {2,3}
  const int l    = lane & 15;
  const int row  = blockIdx.x * 16 + l;    // A row feeding this lane's frag
  const int Mp   = CT * 16;                // padded B stride

  v8f c[CT];
#pragma unroll
  for (int t = 0; t < CT; ++t) c[t] = (v8f){};

  const float* arow = A + (size_t)row * K;
  for (int k0 = 0; k0 < K; k0 += 4) {
    const int ka = k0 + half * 2;
    v2f a;                                  // contiguous -> global_load_b64
    a.x = arow[ka];
    a.y = arow[ka + 1];
    const float* brow = B + (size_t)ka * Mp;
#pragma unroll
    for (int t = 0; t < CT; ++t) {
      v2f b;
      b.x = brow[t * 16 + l];
      b.y = brow[Mp + t * 16 + l];
      c[t] = __builtin_amdgcn_wmma_f32_16x16x4_f32(false, a, false, b,
                                                   (short)0, c[t], false, false);
    }
  }
  // C/D layout: VGPR r -> row (r + 8*half), col = lane&15 within tile
  const int rbase = blockIdx.x * 16 + half * 8;
#pragma unroll
  for (int t = 0; t < CT; ++t) {
    const int col = t * 16 + l;
    if (col < M) {
#pragma unroll
      for (int r = 0; r < 8; ++r)
        C[(size_t)(rbase + r) * M + col] = c[t][r];
    }
  }
}

// Zero-pad W [K,M] -> Wp [K,Mp] so the GEMM loop needs no column guards.
__global__ void gat_pad_w(const float* __restrict__ W, float* __restrict__ Wp,
                          int K, int M, int Mp) {
  int i = blockIdx.x * blockDim.x + threadIdx.x;
  if (i >= K * Mp) return;
  const int k = i / Mp, m = i - k * Mp;
  Wp[i] = (m < M) ? W[k * M + m] : 0.0f;
}

// ---------------------------------------------------------------------------
// el[n,h] = <feat[n,h,:], al[h,:]> ; er likewise. One thread per (n,h).
// ---------------------------------------------------------------------------
__global__ void gat_el_er(const float* __restrict__ feat,
                          const float* __restrict__ al,
                          const float* __restrict__ ar,
                          float* __restrict__ el, float* __restrict__ er,
                          int NH, int H, int D) {
  int i = blockIdx.x * blockDim.x + threadIdx.x;
  if (i >= NH) return;
  const int h = i % H;
  const float* f = feat + (size_t)i * D;       // i == n*H + h
  float sl = 0.f, sr = 0.f;
  for (int d = 0; d < D; ++d) {
    const float v = f[d];
    sl = fmaf(v, al[h * D + d], sl);
    sr = fmaf(v, ar[h * D + d], sr);
  }
  el[i] = sl;
  er[i] = sr;
}

// Monotone float <-> uint mapping so atomicMax(u32) == float max.
__device__ __forceinline__ unsigned ord_f32(float f) {
  unsigned u = __float_as_uint(f);
  return (u & 0x80000000u) ? ~u : (u | 0x80000000u);
}
__device__ __forceinline__ float unord_f32(unsigned u) {
  return __uint_as_float((u & 0x80000000u) ? (u & 0x7FFFFFFFu) : ~u);
}

// Pass 1: edge score + segment max (global_atomic_max_u32 per (dst,h)).
__global__ void gat_edge_max(const int* __restrict__ src,
                             const int* __restrict__ dst,
                             const float* __restrict__ el,
                             const float* __restrict__ er,
                             unsigned* __restrict__ mord,
                             float* __restrict__ ebuf, int EH, int H) {
  int i = blockIdx.x * blockDim.x + threadIdx.x;
  if (i >= EH) return;
  const int e = i / H, h = i - e * H;
  const int s = src[e], d = dst[e];
  float v = el[s * H + h] + er[d * H + h];
  v = (v > 0.f) ? v : 0.2f * v;                // leaky_relu, slope 0.2
  ebuf[i] = v;
  atomicMax(&mord[d * H + h], ord_f32(v));
}

// Pass 2: ex = exp(e - m[dst]); segment sum via global_atomic_add_f32.
__global__ void gat_edge_expsum(const int* __restrict__ dst,
                                const unsigned* __restrict__ mord,
                                float* __restrict__ ebuf,
                                float* __restrict__ ssum, int EH, int H) {
  int i = blockIdx.x * blockDim.x + threadIdx.x;
  if (i >= EH) return;
  const int e = i / H, h = i - e * H;
  const int d = dst[e];
  const float ex = expf(ebuf[i] - unord_f32(mord[d * H + h]));
  ebuf[i] = ex;
  atomicAdd(&ssum[d * H + h], ex);
}

// Pass 3: acc[dst,h,:] += feat[src,h,:] * alpha. One wave32 per (edge,head);
// lanes stride D -> coalesced 128B gathers + f32 atomics that live in L2.
__global__ void gat_edge_agg(const int* __restrict__ src,
                             const int* __restrict__ dst,
                             const float* __restrict__ feat,
                             const float* __restrict__ ebuf,
                             const float* __restrict__ ssum,
                             float* __restrict__ acc, int EH, int H, int D) {
  const int wave = (int)((blockIdx.x * blockDim.x + threadIdx.x) >> 5);
  const int lane = threadIdx.x & 31;
  if (wave >= EH) return;
  const int e = wave / H, h = wave - e * H;
  const int sN = src[e], dN = dst[e];
  const float alpha = ebuf[wave] / ssum[dN * H + h];
  const float* fs = feat + ((size_t)sN * H + h) * D;
  float* ad = acc + ((size_t)dN * H + h) * D;
  for (int d = lane; d < D; d += 32)
    atomicAdd(&ad[d], fs[d] * alpha);
}

// bias + optional ELU + optional residual.
__global__ void gat_epilogue(const float* __restrict__ acc,
                             const float* __restrict__ bias,
                             const float* __restrict__ resid,
                             float* __restrict__ out, int NM, int M, int elu) {
  int i = blockIdx.x * blockDim.x + threadIdx.x;
  if (i >= NM) return;
  const int f = i % M;
  float v = acc[i] + bias[f];
  if (elu) v = (v > 0.f) ? v : (expf(v) - 1.f);
  if (resid) v += resid[i];
  out[i] = v;
}

// log_softmax over C classes, one thread per node (C=40; L2-hot sweeps).
__global__ void gat_logsoftmax(const float* __restrict__ logits,
                               float* __restrict__ out, int n, int C) {
  int i = blockIdx.x * blockDim.x + threadIdx.x;
  if (i >= n) return;
  const float* x = logits + (size_t)i * C;
  float m = -INFINITY;
  for (int c = 0; c < C; ++c) m = fmaxf(m, x[c]);
  float s = 0.f;
  for (int c = 0; c < C; ++c) s += expf(x[c] - m);
  const float ls = logf(s);
  float* o = out + (size_t)i * C;
  for (int c = 0; c < C; ++c) o[c] = x[c] - m - ls;
}

// ---------------------------------------------------------------------------
static void run_layer(const float* h_in, const float* Bmat, const float* al,
                      const float* ar, const float* bias, const int* src,
                      const int* dst, int Fin, int H, int D, bool elu,
                      const float* resid, float* h_out,
                      float* feat, float* el, float* er, unsigned* mord,
                      float* ssum, float* exbuf, float* acc,
                      hipStream_t stream) {
  const int M  = H * D;
  const int NH = GN * H;
  const int EH = GE * H;

  if (M == 128)
    gat_gemm_wmma<8><<<GN / 16, 32, 0, stream>>>(h_in, Bmat, feat, Fin, M);
  else
    gat_gemm_wmma<3><<<GN / 16, 32, 0, stream>>>(h_in, Bmat, feat, Fin, M);

  gat_el_er<<<(NH + 255) / 256, 256, 0, stream>>>(feat, al, ar, el, er, NH, H, D);

  hipMemsetAsync(mord, 0, (size_t)NH * sizeof(unsigned), stream); // -inf in ord space
  hipMemsetAsync(ssum, 0, (size_t)NH * sizeof(float), stream);
  hipMemsetAsync(acc, 0, (size_t)GN * M * sizeof(float), stream);

  gat_edge_max<<<(EH + 255) / 256, 256, 0, stream>>>(src, dst, el, er, mord, exbuf, EH, H);
  gat_edge_expsum<<<(EH + 255) / 256, 256, 0, stream>>>(dst, mord, exbuf, ssum, EH, H);

  const int nblk = (EH + 3) / 4;                     // 4 wave32 per 128-thread block
  gat_edge_agg<<<nblk, 128, 0, stream>>>(src, dst, feat, exbuf, ssum, acc, EH, H, D);

  const int NM = GN * M;
  gat_epilogue<<<(NM + 255) / 256, 256, 0, stream>>>(acc, bias, resid, h_out, NM, M, elu ? 1 : 0);
}

extern "C" void kernel_launch(void* const* d_in, const int* in_sizes, int n_in,
                              void* d_out, int out_size, void* d_ws, size_t ws_size,
                              hipStream_t stream) {
  const float* x   = (const float*)d_in[0];
  const int*   src = (const int*)d_in[1];
  const int*   dst = (const int*)d_in[2];
  const float* W0  = (const float*)d_in[3];
  const float* al0 = (const float*)d_in[4];
  const float* ar0 = (const float*)d_in[5];
  const float* b0  = (const float*)d_in[6];
  const float* W1  = (const float*)d_in[7];
  const float* al1 = (const float*)d_in[8];
  const float* ar1 = (const float*)d_in[9];
  const float* b1  = (const float*)d_in[10];
  const float* W2  = (const float*)d_in[11];
  const float* al2 = (const float*)d_in[12];
  const float* ar2 = (const float*)d_in[13];
  const float* b2  = (const float*)d_in[14];

  // workspace carve-up (floats)
  float* w     = (float*)d_ws;
  float* feat  = w;                              // [N,128]
  float* hA    = feat + (size_t)GN * 128;        // [N,128]
  float* hB    = hA   + (size_t)GN * 128;        // [N,128]
  float* acc   = hB   + (size_t)GN * 128;        // [N,128]
  float* el    = acc  + (size_t)GN * 128;        // [N,2]
  float* er    = el   + (size_t)GN * 2;          // [N,2]
  float* ssum  = er   + (size_t)GN * 2;          // [N,2]
  unsigned* mo = (unsigned*)(ssum + (size_t)GN * 2); // [N,2]
  float* exbuf = (float*)(mo + (size_t)GN * 2);  // [E,2]
  float* W2p   = exbuf + (size_t)GE * 2;         // [128,48] zero-padded W2

  // Zero-pad W2 (128x40 -> 128x48) so the WMMA loop is divergence-free.
  gat_pad_w<<<(FIN * 48 + 255) / 256, 256, 0, stream>>>(W2, W2p, FIN, OUTC, 48);

  // Layer 0: 128 -> 2x64, ELU, residual with x
  run_layer(x,  W0,  al0, ar0, b0, src, dst, FIN,     2, HID,  true,  x,  hA,
            feat, el, er, mo, ssum, exbuf, acc, stream);
  // Layer 1: 128 -> 2x64, ELU, residual with hA
  run_layer(hA, W1,  al1, ar1, b1, src, dst, 2 * HID, 2, HID,  true,  hA, hB,
            feat, el, er, mo, ssum, exbuf, acc, stream);
  // Layer 2: 128 -> 1x40 (padded B), no activation, no residual
  run_layer(hB, W2p, al2, ar2, b2, src, dst, 2 * HID, 1, OUTC, false, nullptr, hA,
            feat, el, er, mo, ssum, exbuf, acc, stream);

  gat_logsoftmax<<<(GN + 255) / 256, 256, 0, stream>>>(hA, (float*)d_out, GN, OUTC);
}